// GeneralOGBConv_36000415875684
// MI455X (gfx1250) — compile-verified
//
#include <hip/hip_runtime.h>
#include <hip/hip_bf16.h>

typedef __attribute__((ext_vector_type(2))) float v2f;
typedef __attribute__((ext_vector_type(8))) float v8f;

#define CCH 128  // channel count

// ---------------------------------------------------------------- init ------
__global__ void init_zero_kernel(float* __restrict__ out, float* __restrict__ deg,
                                 int n_out, int n_deg) {
  int i = blockIdx.x * blockDim.x + threadIdx.x;
  if (i < n_out) out[i] = 0.0f;
  if (i < n_deg) deg[i] = 0.0f;
}

// ------------------------------------------------------------- degree -------
__global__ void deg_kernel(const int* __restrict__ edge_index, float* __restrict__ deg,
                           int E) {
  int e = blockIdx.x * blockDim.x + threadIdx.x;
  if (e < E) atomicAdd(&deg[edge_index[e]], 1.0f);  // row = edge_index[0][e]
}

__global__ void dinv_kernel(float* __restrict__ deg, int n) {
  int i = blockIdx.x * blockDim.x + threadIdx.x;
  if (i < n) {
    float d = deg[i];
    deg[i] = (d > 0.0f) ? rsqrtf(d) : 0.0f;  // in-place deg -> dinv
  }
}

// ------------------------------------------------- xw = x @ weight (WMMA) ---
// One wave computes a 16-row strip of xw across all 128 columns:
// 8 accumulators of v8f (16x16 f32 tiles), K stepped by 4 via
// V_WMMA_F32_16X16X4_F32. x is streamed exactly once; weight (64KB) is
// cache-resident.
__global__ void gemm_wmma_kernel(const float* __restrict__ x, const float* __restrict__ w,
                                 float* __restrict__ xw, int n_mtiles, int n_nodes) {
  const int lane  = threadIdx.x & 31;
  const int wid   = threadIdx.x >> 5;
  const int mtile = blockIdx.x * (blockDim.x >> 5) + wid;
  if (mtile >= n_mtiles) return;  // wave-uniform exit: EXEC stays all-1s for WMMA

  const int half = lane >> 4;   // 0: lanes 0-15, 1: lanes 16-31
  const int lrow = lane & 15;

  const int m = mtile * 16 + lrow;                      // A-matrix row (x row)
  const int mload = (m < n_nodes) ? m : (n_nodes - 1);  // clamp loads only
  const float* __restrict__ xrow = x + (size_t)mload * CCH;

  v8f acc[8];
#pragma unroll
  for (int nt = 0; nt < 8; ++nt) acc[nt] = (v8f)0.0f;

  for (int kk = 0; kk < CCH; kk += 4) {
    const int kb = kk + 2 * half;  // K base for this half-wave
    v2f a;
    a[0] = xrow[kb];
    a[1] = xrow[kb + 1];
#pragma unroll
    for (int nt = 0; nt < 8; ++nt) {
      const int n = nt * 16 + lrow;
      v2f b;
      b[0] = w[(size_t)kb * CCH + n];
      b[1] = w[(size_t)(kb + 1) * CCH + n];
      // (neg_a, A, neg_b, B, c_mod, C, reuse_a, reuse_b)
      acc[nt] = __builtin_amdgcn_wmma_f32_16x16x4_f32(
          false, a, false, b, (short)0, acc[nt], false, false);
    }
  }

  // D layout: VGPR v -> row (v + 8*half), col = lane%16 within the 16x16 tile
  if (mtile * 16 + 16 <= n_nodes) {
    // fast path (always taken when N % 16 == 0): no per-lane guards
#pragma unroll
    for (int nt = 0; nt < 8; ++nt) {
      const int n = nt * 16 + lrow;
#pragma unroll
      for (int v = 0; v < 8; ++v) {
        const int r = mtile * 16 + v + 8 * half;
        xw[(size_t)r * CCH + n] = acc[nt][v];
      }
    }
  } else {
#pragma unroll
    for (int nt = 0; nt < 8; ++nt) {
      const int n = nt * 16 + lrow;
#pragma unroll
      for (int v = 0; v < 8; ++v) {
        const int r = mtile * 16 + v + 8 * half;
        if (r < n_nodes) xw[(size_t)r * CCH + n] = acc[nt][v];
      }
    }
  }
}

// -------------------------------------------- edge message + scatter-add ----
// One WAVE per edge, 4 channels per lane (32 lanes x 4 = 128 channels).
// Gathers become global_load_b128 (512B/edge fully coalesced), edge scalars
// are fetched once per wave, and the irreducible 128 f32 atomics/edge execute
// as cacheline-granular RMW in the (192MB, fully resident) L2.
__global__ void scatter_kernel(const float* __restrict__ xw, const float* __restrict__ dinv,
                               const int* __restrict__ edge_index,
                               const int* __restrict__ edge_feature,
                               const float* __restrict__ emb0, const float* __restrict__ emb1,
                               const float* __restrict__ emb2,
                               float* __restrict__ out, int E) {
  const int lane = threadIdx.x & 31;
  const int e = blockIdx.x * (blockDim.x >> 5) + (threadIdx.x >> 5);
  if (e >= E) return;  // wave-uniform

  const int row = edge_index[e];          // source
  const int col = edge_index[E + e];      // target
  const int f0 = edge_feature[3 * e + 0];
  const int f1 = edge_feature[3 * e + 1];
  const int f2 = edge_feature[3 * e + 2];

  const float norm = dinv[row] * dinv[col];

  const float4 xv = *((const float4*)(xw + (size_t)row * CCH) + lane);
  const float4 a0 = *((const float4*)(emb0 + (size_t)f0 * CCH) + lane);
  const float4 a1 = *((const float4*)(emb1 + (size_t)f1 * CCH) + lane);
  const float4 a2 = *((const float4*)(emb2 + (size_t)f2 * CCH) + lane);

  float m[4];
  m[0] = norm * (xv.x + a0.x + a1.x + a2.x);
  m[1] = norm * (xv.y + a0.y + a1.y + a2.y);
  m[2] = norm * (xv.z + a0.z + a1.z + a2.z);
  m[3] = norm * (xv.w + a0.w + a1.w + a2.w);

  float* dst = out + (size_t)col * CCH + 4 * lane;
#pragma unroll
  for (int j = 0; j < 4; ++j) atomicAdd(dst + j, m[j]);
}

// ---------------------------------------------------------------- launch ----
extern "C" void kernel_launch(void* const* d_in, const int* in_sizes, int n_in,
                              void* d_out, int out_size, void* d_ws, size_t ws_size,
                              hipStream_t stream) {
  const float* x            = (const float*)d_in[0];
  const int*   edge_index   = (const int*)d_in[1];
  const int*   edge_feature = (const int*)d_in[2];
  const float* weight       = (const float*)d_in[3];
  const float* emb0         = (const float*)d_in[4];
  const float* emb1         = (const float*)d_in[5];
  const float* emb2         = (const float*)d_in[6];
  float*       out          = (float*)d_out;

  const int N = in_sizes[0] / CCH;   // 50000 nodes
  const int E = in_sizes[1] / 2;     // 625000 edges

  // workspace layout: xw [N*C f32] | deg/dinv [N f32]
  float* xw  = (float*)d_ws;
  float* deg = xw + (size_t)N * CCH;

  {
    const int total = N * CCH;  // covers both out (N*C) and deg (N)
    init_zero_kernel<<<(total + 255) / 256, 256, 0, stream>>>(out, deg, total, N);
  }

  deg_kernel<<<(E + 255) / 256, 256, 0, stream>>>(edge_index, deg, E);
  dinv_kernel<<<(N + 255) / 256, 256, 0, stream>>>(deg, N);

  {
    const int mtiles = (N + 15) / 16;            // 3125
    const int wavesPerBlock = 8;                 // 256 threads = 8 waves
    const int blocks = (mtiles + wavesPerBlock - 1) / wavesPerBlock;
    gemm_wmma_kernel<<<blocks, wavesPerBlock * 32, 0, stream>>>(x, weight, xw, mtiles, N);
  }

  {
    const int wavesPerBlock = 8;                 // 1 edge per wave
    const int blocks = (E + wavesPerBlock - 1) / wavesPerBlock;
    scatter_kernel<<<blocks, wavesPerBlock * 32, 0, stream>>>(
        xw, deg, edge_index, edge_feature, emb0, emb1, emb2, out, E);
  }
}